// calc_padding_56075093016978
// MI455X (gfx1250) — compile-verified
//
#include <hip/hip_runtime.h>

typedef __attribute__((ext_vector_type(2))) float v2f;
typedef __attribute__((ext_vector_type(4))) float v4f;
typedef __attribute__((ext_vector_type(8))) float v8f;

#define HIN   224
#define WIN   224
#define HP    226
#define WP2   226
#define PLANE (HP * WP2)        // 51076 floats, divisible by 4; plane stride 16B-aligned
#define PLANE4 (PLANE / 4)      // 12769 float4s exactly

// ---------------------------------------------------------------------------
// Kernel 1: zero-pad copy, b128 stores. One plane per blockIdx.y; flat float4
// index over the padded plane. ~830 MB traffic -> ~36 us at 23.3 TB/s.
// ---------------------------------------------------------------------------
__global__ __launch_bounds__(256) void pad_copy_kernel(const float* __restrict__ x,
                                                       float* __restrict__ xp) {
  const int q = blockIdx.x * 256 + threadIdx.x;     // float4 index within plane
  if (q >= PLANE4) return;
  const long long bc = blockIdx.y;                  // 0..B*C-1
  const float* __restrict__ xs = x + bc * (long long)(HIN * WIN);

  v4f v;
  #pragma unroll
  for (int e = 0; e < 4; ++e) {
    const int idx = q * 4 + e;
    const int h = idx / WP2;                        // const-div -> mulhi
    const int w = idx - h * WP2;
    float t = 0.0f;
    if (h >= 1 && h <= HIN && w >= 1 && w <= WIN)
      t = xs[(h - 1) * WIN + (w - 1)];
    v[e] = t;
  }
  *(v4f*)(xp + bc * (long long)PLANE + (long long)q * 4) = v;   // global_store_b128
}

// ---------------------------------------------------------------------------
// Kernel 2: border MLP via V_WMMA_F32_16X16X4_F32 (exact f32 matrix pipe).
// One wave32 handles 16 border positions of one (plane, branch).
//   layer1: [8x6(+2 zero-pad)] x [6x16]  -> 2x wmma K=4
//   layer2: [8x8] x [8x16]               -> 2x wmma K=4 (D->B via shuffles)
//   layer3: 8->1 dot + ReLU in VALU, lanes 0..15 store.
// All masking via selects on unconditionally-executed loads, so no divergent
// branches before the WMMAs (EXEC all-ones as required).
// ---------------------------------------------------------------------------
__global__ __launch_bounds__(32) void border_mlp_kernel(
    float* __restrict__ xp,
    const float* __restrict__ w1A, const float* __restrict__ b1A,
    const float* __restrict__ w2A, const float* __restrict__ b2A,
    const float* __restrict__ w3A, const float* __restrict__ b3A,
    const float* __restrict__ w1B, const float* __restrict__ b1B,
    const float* __restrict__ w2B, const float* __restrict__ b2B,
    const float* __restrict__ w3B, const float* __restrict__ b3B,
    long long inBaseA, long long inBaseB,
    long long outBaseA, long long outBaseB,
    long long strideR, long long strideP, long long outStride)
{
  const int L  = threadIdx.x;
  const int m  = L & 15;   // M (A rows) / N (positions) index
  const int hi = L >> 4;   // wave half -> which K pair

  const int branch = blockIdx.z;
  const float* w1 = branch ? w1B : w1A;
  const float* b1 = branch ? b1B : b1A;
  const float* w2 = branch ? w2B : w2A;
  const float* b2 = branch ? b2B : b2A;
  const float* w3 = branch ? w3B : w3A;
  const float* b3 = branch ? b3B : b3A;
  const long long inBase  = branch ? inBaseB  : inBaseA;
  const long long outBase = branch ? outBaseB : outBaseA;

  float* plane = xp + (long long)blockIdx.y * PLANE;
  const int cc = blockIdx.x * 16 + m + 1;   // padded output coord 1..224

  // ---- layer 1 operands (A = W1 8x6 zero-padded to 16x8, B = input patch) --
  auto w1get = [&](int k) -> float {                     // A[m][k], k global 0..7
    float t = w1[(m & 7) * 6 + (k < 6 ? k : 0)];         // always-safe load
    return (m < 8 && k < 6) ? t : 0.0f;                  // then mask
  };
  auto pget = [&](int k) -> float {                      // B[k][n=m], k global 0..7
    int kk = (k < 6) ? k : 0;
    int r  = (kk >= 3) ? 1 : 0;                          // window row (REF0)
    int c  = kk - 3 * r;                                 // window col (WP)
    float t = plane[inBase + (long long)r * strideR +
                    (long long)(cc - 1 + c) * strideP];
    return (k < 6) ? t : 0.0f;
  };

  v2f a0, a1, p0, p1;
  a0[0] = w1get(hi * 2);     a0[1] = w1get(hi * 2 + 1);
  a1[0] = w1get(4 + hi * 2); a1[1] = w1get(5 + hi * 2);
  p0[0] = pget(hi * 2);      p0[1] = pget(hi * 2 + 1);
  p1[0] = pget(4 + hi * 2);  p1[1] = pget(5 + hi * 2);

  v8f acc;                                   // C layout: vgpr r holds M=r / M=r+8
  #pragma unroll
  for (int r = 0; r < 8; ++r) {
    float t = b1[r];                         // unconditional uniform load
    acc[r] = hi ? 0.0f : t;                  // cndmask, no exec branching
  }

  acc = __builtin_amdgcn_wmma_f32_16x16x4_f32(false, a0, false, p0, (short)0, acc, false, false);
  acc = __builtin_amdgcn_wmma_f32_16x16x4_f32(false, a1, false, p1, (short)0, acc, false, false);
  #pragma unroll
  for (int r = 0; r < 8; ++r) acc[r] = fmaxf(acc[r], 0.0f);   // ReLU

  // ---- layer 2: A = W2 (8x8), B = relu(y1) converted D-layout -> B-layout --
  auto w2get = [&](int k) -> float {
    float t = w2[(m & 7) * 8 + k];
    return (m < 8) ? t : 0.0f;
  };
  v2f a20, a21;
  a20[0] = w2get(hi * 2);     a20[1] = w2get(hi * 2 + 1);
  a21[0] = w2get(4 + hi * 2); a21[1] = w2get(5 + hi * 2);

  // y1[h][j] lives in acc[h], lanes 0-15 (j = lane). B[k][n]: vgpr=k&1, lane=n+16*(k>>1).
  v2f y0, y1;
  {
    float l0 = __shfl(acc[0], m, 32), h0 = __shfl(acc[2], m, 32);
    float l1 = __shfl(acc[1], m, 32), h1 = __shfl(acc[3], m, 32);
    float l2 = __shfl(acc[4], m, 32), h2 = __shfl(acc[6], m, 32);
    float l3 = __shfl(acc[5], m, 32), h3 = __shfl(acc[7], m, 32);
    y0[0] = hi ? h0 : l0;
    y0[1] = hi ? h1 : l1;
    y1[0] = hi ? h2 : l2;
    y1[1] = hi ? h3 : l3;
  }

  v8f acc2;
  #pragma unroll
  for (int r = 0; r < 8; ++r) {
    float t = b2[r];                         // unconditional uniform load
    acc2[r] = hi ? 0.0f : t;
  }
  acc2 = __builtin_amdgcn_wmma_f32_16x16x4_f32(false, a20, false, y0, (short)0, acc2, false, false);
  acc2 = __builtin_amdgcn_wmma_f32_16x16x4_f32(false, a21, false, y1, (short)0, acc2, false, false);
  #pragma unroll
  for (int r = 0; r < 8; ++r) acc2[r] = fmaxf(acc2[r], 0.0f); // ReLU

  // ---- layer 3: 8 -> 1 dot per position, final ReLU ----
  float o = b3[0];
  #pragma unroll
  for (int r = 0; r < 8; ++r) o += w3[r] * acc2[r];
  o = fmaxf(o, 0.0f);

  if (hi == 0)                                // lanes 0..15 hold positions j=0..15
    plane[outBase + (long long)cc * outStride] = o;
}

// ---------------------------------------------------------------------------
// Launcher. Input order: x, then (w1,b1,w2,b2,w3,b3) for t, b, l, r.
// Kernel order enforces the reference's ring dependence:
//   pad/copy -> top+bottom rows -> left+right cols (which read the corners).
// ---------------------------------------------------------------------------
extern "C" void kernel_launch(void* const* d_in, const int* in_sizes, int n_in,
                              void* d_out, int out_size, void* d_ws, size_t ws_size,
                              hipStream_t stream) {
  (void)n_in; (void)out_size; (void)d_ws; (void)ws_size;
  const float* x = (const float*)d_in[0];
  float* xp = (float*)d_out;
  const int BC = in_sizes[0] / (HIN * WIN);   // 32*64 = 2048 planes

#define WPTR(i) ((const float*)d_in[(i)])
  // t: 1..6, b: 7..12, l: 13..18, r: 19..24
  pad_copy_kernel<<<dim3((PLANE4 + 255) / 256, BC, 1), 256, 0, stream>>>(x, xp);

  // top (branch A) + bottom (branch B): scan along columns (strideP = 1)
  border_mlp_kernel<<<dim3(WIN / 16, BC, 2), 32, 0, stream>>>(
      xp,
      WPTR(1),  WPTR(2),  WPTR(3),  WPTR(4),  WPTR(5),  WPTR(6),
      WPTR(7),  WPTR(8),  WPTR(9),  WPTR(10), WPTR(11), WPTR(12),
      /*inBaseA  (rows 1,2)   */ (long long)1 * WP2,
      /*inBaseB  (rows 223,4) */ (long long)223 * WP2,
      /*outBaseA (row 0)      */ 0LL,
      /*outBaseB (row 225)    */ (long long)225 * WP2,
      /*strideR*/ (long long)WP2, /*strideP*/ 1LL, /*outStride*/ 1LL);

  // left (branch A) + right (branch B): scan along rows (strideP = 226)
  border_mlp_kernel<<<dim3(HIN / 16, BC, 2), 32, 0, stream>>>(
      xp,
      WPTR(13), WPTR(14), WPTR(15), WPTR(16), WPTR(17), WPTR(18),
      WPTR(19), WPTR(20), WPTR(21), WPTR(22), WPTR(23), WPTR(24),
      /*inBaseA  (cols 1,2)   */ 1LL,
      /*inBaseB  (cols 223,4) */ 223LL,
      /*outBaseA (col 0)      */ 0LL,
      /*outBaseB (col 225)    */ 225LL,
      /*strideR*/ 1LL, /*strideP*/ (long long)WP2, /*outStride*/ (long long)WP2);
#undef WPTR
}